// LRCoulomb_NB_47991964566168
// MI455X (gfx1250) — compile-verified
//
#include <hip/hip_runtime.h>
#include <hip/hip_bf16.h>

#ifndef __has_builtin
#define __has_builtin(x) 0
#endif

// gfx1250 async global->LDS copy (ASYNCcnt-tracked), probe-verified to lower to
// global_load_async_to_lds_b64 + s_wait_asynccnt on this toolchain.
#if __has_builtin(__builtin_amdgcn_global_load_async_to_lds_b64) && \
    __has_builtin(__builtin_amdgcn_s_wait_asynccnt)
#define USE_ASYNC_LDS 1
#else
#define USE_ASYNC_LDS 0
#endif

#define RC_F        4.6f
#define COUL_CONST  7.1998226f

typedef int v2i __attribute__((vector_size(2 * sizeof(int))));
typedef __attribute__((address_space(1))) v2i gbl_v2i_t;   // global source
typedef __attribute__((address_space(3))) v2i lds_v2i_t;   // LDS destination

__global__ __launch_bounds__(256) void lrcoulomb_nb_kernel(
    const float* __restrict__ coord,   // [N,3]
    const float* __restrict__ q,       // [N]
    const int*   __restrict__ idx,     // [N,64]
    const int*   __restrict__ msk,     // [N,64] (0/1)
    double*      __restrict__ out,     // [N] float64
    int N)
{
    constexpr int M     = 64;
    constexpr int WAVES = 8;                    // 256 threads / wave32

    __shared__ __align__(16) int s_idx[WAVES][M];
    __shared__ __align__(16) int s_msk[WAVES][M];

    const int lane = threadIdx.x & 31;
    const int wave = threadIdx.x >> 5;
    const int i    = blockIdx.x * WAVES + wave; // one atom per wave
    if (i >= N) return;                         // uniform per wave

    // ---- Issue async staging of this wave's idx/mask row (coalesced b64/lane) ----
    const int base = i * M + lane * 2;          // 8B aligned
#if USE_ASYNC_LDS
    __builtin_amdgcn_global_load_async_to_lds_b64(
        (gbl_v2i_t*)(idx + base), (lds_v2i_t*)&s_idx[wave][lane * 2], 0, 0);
    __builtin_amdgcn_global_load_async_to_lds_b64(
        (gbl_v2i_t*)(msk + base), (lds_v2i_t*)&s_msk[wave][lane * 2], 0, 0);
#else
    {
        const int2 vi = *(const int2*)(idx + base);
        const int2 vm = *(const int2*)(msk + base);
        s_idx[wave][lane * 2]     = vi.x;
        s_idx[wave][lane * 2 + 1] = vi.y;
        s_msk[wave][lane * 2]     = vm.x;
        s_msk[wave][lane * 2 + 1] = vm.y;
    }
#endif

    // ---- Overlap: fetch self-atom data while the async DMA is in flight ----
    // (LOADcnt-tracked, independent of ASYNCcnt -> latency hidden for free)
    const float xi = coord[3 * i + 0];
    const float yi = coord[3 * i + 1];
    const float zi = coord[3 * i + 2];
    const float qi = q[i];

    const float inv_rc = 1.0f / RC_F;
    const float x_clip = 1.0f - 1e-7f;

#if USE_ASYNC_LDS
    __builtin_amdgcn_s_wait_asynccnt(0);        // per-wave counter: no barrier needed
#endif

    double acc = 0.0;                           // f64 accumulation of f32 terms

#pragma unroll
    for (int h = 0; h < 2; ++h) {
        const int m  = lane + 32 * h;
        const int j  = s_idx[wave][m];
        const int pm = s_msk[wave][m];

        const float* cj = coord + 3 * j;        // L2-resident random gather (b96)
        const float dx = cj[0] - xi;
        const float dy = cj[1] - yi;
        const float dz = cj[2] - zi;

        float d2 = dx * dx + dy * dy + dz * dz;
        if (pm) d2 = 1.0f;                      // padded pair -> d = 1
        const float d  = sqrtf(d2);
        const float qj = pm ? 0.0f : q[j];

        // exp_cutoff(d, rc) = d<rc ? exp(1 - 1/(1-x)), x = clip((d/rc)^2, 0, 1-1e-7)
        const float t = d * inv_rc;
        float x = t * t;
        x = fminf(x, x_clip);
        const float fcut = (d < RC_F) ? expf(1.0f - 1.0f / (1.0f - x)) : 0.0f;
        const float fc   = 1.0f - fcut;

        const float term = COUL_CONST * fc * (qi * qj) / d;
        acc += (double)term;
    }

    // ---- wave32 butterfly reduction in double ----
#pragma unroll
    for (int off = 16; off > 0; off >>= 1)
        acc += __shfl_down(acc, off, 32);

    if (lane == 0) out[i] = acc;
}

extern "C" void kernel_launch(void* const* d_in, const int* in_sizes, int n_in,
                              void* d_out, int out_size, void* d_ws, size_t ws_size,
                              hipStream_t stream) {
    // setup_inputs() order: coord [N,3] f32, q [N] f32, idx_j_coul [N,64] int,
    // nb_pad_mask_coul [N,64] int(0/1). Output: e_h [N] float64.
    const float* coord = (const float*)d_in[0];
    const float* q     = (const float*)d_in[1];
    const int*   idx   = (const int*)d_in[2];
    const int*   msk   = (const int*)d_in[3];
    double*      out   = (double*)d_out;

    const int N = in_sizes[1];                  // q has N elements
    const int waves_per_block = 8;              // 256 threads, wave32
    const int blocks = (N + waves_per_block - 1) / waves_per_block;

    lrcoulomb_nb_kernel<<<blocks, 256, 0, stream>>>(coord, q, idx, msk, out, N);
}